// Aggregator_15126874816994
// MI455X (gfx1250) — compile-verified
//
#include <hip/hip_runtime.h>
#include <hip/hip_bf16.h>
#include <math.h>

typedef __attribute__((ext_vector_type(2))) float v2f;
typedef __attribute__((ext_vector_type(8))) float v8f;

#define B_ 512
#define E_ 64
#define N_ 32
#define D_ 128

// LDS row strides padded to 132 floats: 132 % 64 == 4, so the 16-lane
// column reads in the WMMA phase hit 16 distinct banks (4*f + k), and
// 132*4 = 528 bytes keeps float4/float2 accesses naturally aligned.
#define XS_STRIDE 132
#define WL_STRIDE 132

// gfx1250 has a hardware V_TANH_F32 transcendental; use it if the builtin
// is declared, otherwise fall back to libm tanhf (branchy ocml expansion).
#if defined(__has_builtin)
#if __has_builtin(__builtin_amdgcn_tanhf)
#define FAST_TANH(x) __builtin_amdgcn_tanhf(x)
#endif
#endif
#ifndef FAST_TANH
#define FAST_TANH(x) tanhf(x)
#endif

__global__ __launch_bounds__(256)
void aggregator_kernel(const float* __restrict__ self_v,
                       const float* __restrict__ nbr_v,
                       const float* __restrict__ nbr_r,
                       const float* __restrict__ user_e,
                       const float* __restrict__ W,
                       const float* __restrict__ bias,
                       float* __restrict__ out)
{
    __shared__ float xs[16 * XS_STRIDE];    //  8.25 KB: X = self + agg
    __shared__ float Wl[128 * WL_STRIDE];   // 66 KB: staged W

    const int tid  = threadIdx.x;
    const int wave = tid >> 5;          // 0..7
    const int lane = tid & 31;
    const int g0   = blockIdx.x * 16;   // flat b*E + e base; 16 rows per block
    const int bi   = g0 >> 6;           // E == 64, all 16 rows share one batch b

    // ---- stage W (128x128 f32) into padded LDS, coalesced float4 ----
    for (int i = tid; i < 128 * 32; i += 256) {
        const int row = i >> 5;
        const int c4  = (i & 31) << 2;
        const float4 w4 = *(const float4*)(W + row * D_ + c4);
        *(float4*)(&Wl[row * WL_STRIDE + c4]) = w4;
    }

    // ---- phase 1: attention softmax + neighbor aggregation ----
    // Each lane owns 4 contiguous dims: u4 = user[b, lane*4 .. lane*4+3]
    const float4 u4 = *(const float4*)(user_e + (size_t)bi * D_ + lane * 4);

    for (int rr = 0; rr < 2; ++rr) {
        const int r = wave * 2 + rr;                 // row 0..15 in block
        const size_t base = (size_t)(g0 + r) * (N_ * D_);

        // scores[n] = dot(user, rel[n,:]) ; wave streams one 512B row per n
        float myscore = 0.0f;
        #pragma unroll 4
        for (int n = 0; n < N_; ++n) {
            const float4 rl = *(const float4*)(nbr_r + base + n * D_ + lane * 4);
            float p = u4.x * rl.x + u4.y * rl.y + u4.z * rl.z + u4.w * rl.w;
            #pragma unroll
            for (int m = 16; m >= 1; m >>= 1) p += __shfl_xor(p, m, 32);
            if (lane == n) myscore = p;              // lane n keeps score n
        }

        // softmax over 32 neighbors (one score per lane)
        float mx = myscore;
        #pragma unroll
        for (int m = 16; m >= 1; m >>= 1) mx = fmaxf(mx, __shfl_xor(mx, m, 32));
        const float ev = __expf(myscore - mx);
        float s = ev;
        #pragma unroll
        for (int m = 16; m >= 1; m >>= 1) s += __shfl_xor(s, m, 32);
        const float attn = ev / s;

        // agg[d] = sum_n attn[n] * nv[n,d] ; same coalesced streaming
        float ax = 0.f, ay = 0.f, az = 0.f, aw = 0.f;
        #pragma unroll 4
        for (int n = 0; n < N_; ++n) {
            const float a = __shfl(attn, n, 32);
            const float4 v = *(const float4*)(nbr_v + base + n * D_ + lane * 4);
            ax = fmaf(a, v.x, ax); ay = fmaf(a, v.y, ay);
            az = fmaf(a, v.z, az); aw = fmaf(a, v.w, aw);
        }
        const float4 sv = *(const float4*)(self_v + (size_t)(g0 + r) * D_ + lane * 4);
        float4 x4;
        x4.x = ax + sv.x; x4.y = ay + sv.y; x4.z = az + sv.z; x4.w = aw + sv.w;
        *(float4*)(&xs[r * XS_STRIDE + lane * 4]) = x4;
    }
    __syncthreads();

    // ---- phase 2: out[16x128] = X[16x128] @ W^T, fp32 WMMA 16x16x4 ----
    // Wave w computes output columns f0..f0+15.
    const int f0   = wave << 4;
    const int half = lane >> 4;     // 0: K=k0,k0+1   1: K=k0+2,k0+3
    const int l16  = lane & 15;     // M (for A) / N (for B)

    v8f c = {0.f,0.f,0.f,0.f,0.f,0.f,0.f,0.f};
    #pragma unroll 8
    for (int k0 = 0; k0 < D_; k0 += 4) {
        const int ka = k0 + 2 * half;
        // A[m][k] = X[m][k] : lane l16 row, (ka, ka+1)
        const v2f av = *(const v2f*)(&xs[l16 * XS_STRIDE + ka]);
        // B[k][n] = W[f0+n][k] : lane l16 -> W row f0+l16, (ka, ka+1)
        const v2f bv = *(const v2f*)(&Wl[(f0 + l16) * WL_STRIDE + ka]);
        c = __builtin_amdgcn_wmma_f32_16x16x4_f32(false, av, false, bv,
                                                  (short)0, c, false, false);
    }

    // epilogue: + bias, tanh, store. C layout: VGPR j -> M = j + 8*half, N = l16
    const float bb = bias[f0 + l16];
    float* op = out + (size_t)g0 * D_ + f0 + l16;
    #pragma unroll
    for (int j = 0; j < 8; ++j) {
        const int m = j + 8 * half;
        op[(size_t)m * D_] = FAST_TANH(c[j] + bb);
    }
}

extern "C" void kernel_launch(void* const* d_in, const int* in_sizes, int n_in,
                              void* d_out, int out_size, void* d_ws, size_t ws_size,
                              hipStream_t stream) {
    const float* self_v = (const float*)d_in[0];
    const float* nbr_v  = (const float*)d_in[1];
    const float* nbr_r  = (const float*)d_in[2];
    const float* user_e = (const float*)d_in[3];
    const float* W      = (const float*)d_in[4];
    const float* bias   = (const float*)d_in[5];
    float* outp         = (float*)d_out;

    const int blocks = (B_ * E_) / 16;   // 2048 blocks, 16 (b,e) rows each
    aggregator_kernel<<<blocks, 256, 0, stream>>>(self_v, nbr_v, nbr_r,
                                                  user_e, W, bias, outp);
}